// DiscriminativeLoss_72988674228839
// MI455X (gfx1250) — compile-verified
//
#include <hip/hip_runtime.h>
#include <hip/hip_bf16.h>

typedef __attribute__((ext_vector_type(2))) float v2f;
typedef __attribute__((ext_vector_type(8))) float v8f;

#define B_   8
#define D_   16
#define HW_  (512*512)
#define K_   32
#define GAMMA_ 0.001f

// workspace layout (in floats)
#define WS_SUMS   0                       // [B][D][K]  d-major, k innermost
#define WS_CNT    (B_*D_*K_)              // [B][K]
#define WS_VAR    (WS_CNT + B_*K_)        // [B][K]
#define WS_MEANS  (WS_VAR + B_*K_)        // [B][D][K]
#define WS_TOTAL  (WS_MEANS + B_*D_*K_)   // 8704 floats

__device__ __forceinline__ unsigned lds_addr_of(const void* p) {
    // AS3->generic casts on AMDGPU place the 32-bit LDS address in the low word.
    return (unsigned)(uintptr_t)p;
}

__global__ void dl_init(float* __restrict__ ws, float* __restrict__ out) {
    int idx = blockIdx.x * 256 + threadIdx.x;
    if (idx < WS_TOTAL) ws[idx] = 0.f;
    if (idx < 3) out[idx] = 0.f;
}

// Pass 1: per-cluster sums + counts. 4096 pixels per block.
// Labels staged global->LDS with CDNA5 async loads (ASYNCcnt), embedding
// streamed with b128 loads + global_prefetch of the next d-plane.
__global__ void dl_sums(const float* __restrict__ emb, const int* __restrict__ lab,
                        float* __restrict__ ws) {
    __shared__ int4  slab[1024];      // 4096 labels, 16B-aligned for b128
    __shared__ float lsum[D_ * K_];   // [d][k], k innermost -> 32 consecutive banks
    __shared__ float lcnt[K_];
    const int tid = threadIdx.x;
    const int b   = blockIdx.y;
    const int p0  = blockIdx.x * 4096;

    // Kick off async label staging first so it overlaps the LDS zeroing below.
    const int4* lab4 = (const int4*)(lab + (size_t)b * HW_ + p0);
    const unsigned lds0 = lds_addr_of(slab);
#pragma unroll
    for (int i = 0; i < 4; ++i) {
        unsigned la = lds0 + (unsigned)(tid + i * 256) * 16u;
        unsigned long long ga = (unsigned long long)(uintptr_t)(lab4 + tid + i * 256);
        asm volatile("global_load_async_to_lds_b128 %0, %1, off"
                     :: "v"(la), "v"(ga) : "memory");
    }

    for (int i = tid; i < D_ * K_; i += 256) lsum[i] = 0.f;
    if (tid < K_) lcnt[tid] = 0.f;
    __syncthreads();

    // Each lane reads back only the bytes it staged itself.
    asm volatile("s_wait_asynccnt 0x0" ::: "memory");
    int4 L[4];
#pragma unroll
    for (int i = 0; i < 4; ++i) {
        L[i] = slab[tid + i * 256];
        if (L[i].x > 0) atomicAdd(&lcnt[L[i].x - 1], 1.f);
        if (L[i].y > 0) atomicAdd(&lcnt[L[i].y - 1], 1.f);
        if (L[i].z > 0) atomicAdd(&lcnt[L[i].z - 1], 1.f);
        if (L[i].w > 0) atomicAdd(&lcnt[L[i].w - 1], 1.f);
    }

    for (int d = 0; d < D_; ++d) {
        const float4* e4 = (const float4*)(emb + ((size_t)b * D_ + d) * HW_ + p0);
        if (d + 1 < D_) {
            // 256 lanes x 64B stride cover the next plane's 16KB window.
            __builtin_prefetch(emb + ((size_t)b * D_ + d + 1) * HW_ + p0 + tid * 16);
        }
#pragma unroll
        for (int i = 0; i < 4; ++i) {
            float4 v = e4[tid + i * 256];
            if (L[i].x > 0) atomicAdd(&lsum[d * K_ + L[i].x - 1], v.x);
            if (L[i].y > 0) atomicAdd(&lsum[d * K_ + L[i].y - 1], v.y);
            if (L[i].z > 0) atomicAdd(&lsum[d * K_ + L[i].z - 1], v.z);
            if (L[i].w > 0) atomicAdd(&lsum[d * K_ + L[i].w - 1], v.w);
        }
    }
    __syncthreads();
    for (int i = tid; i < D_ * K_; i += 256)
        atomicAdd(&ws[WS_SUMS + b * D_ * K_ + i], lsum[i]);
    if (tid < K_)
        atomicAdd(&ws[WS_CNT + b * K_ + tid], lcnt[tid]);
}

__global__ void dl_means(float* __restrict__ ws) {
    int idx = blockIdx.x * 256 + threadIdx.x;
    if (idx >= B_ * D_ * K_) return;
    int b = idx / (D_ * K_);
    int k = idx % K_;
    float c = ws[WS_CNT + b * K_ + k];
    ws[WS_MEANS + idx] = ws[WS_SUMS + idx] / fmaxf(c, 1.f);
}

// Pass 2: per-pixel relu(||e - mu|| - 0.5)^2 accumulated per cluster.
__global__ void dl_var(const float* __restrict__ emb, const int* __restrict__ lab,
                       float* __restrict__ ws) {
    __shared__ int4  slab[1024];
    __shared__ float lmean[D_ * K_];  // [d][k]
    __shared__ float lvar[K_];
    const int tid = threadIdx.x;
    const int b   = blockIdx.y;
    const int p0  = blockIdx.x * 4096;

    // Async label staging overlaps the means-table load below.
    const int4* lab4 = (const int4*)(lab + (size_t)b * HW_ + p0);
    const unsigned lds0 = lds_addr_of(slab);
#pragma unroll
    for (int i = 0; i < 4; ++i) {
        unsigned la = lds0 + (unsigned)(tid + i * 256) * 16u;
        unsigned long long ga = (unsigned long long)(uintptr_t)(lab4 + tid + i * 256);
        asm volatile("global_load_async_to_lds_b128 %0, %1, off"
                     :: "v"(la), "v"(ga) : "memory");
    }

    for (int i = tid; i < D_ * K_; i += 256) lmean[i] = ws[WS_MEANS + b * D_ * K_ + i];
    if (tid < K_) lvar[tid] = 0.f;
    __syncthreads();

    asm volatile("s_wait_asynccnt 0x0" ::: "memory");
    int4 L[4];
#pragma unroll
    for (int i = 0; i < 4; ++i) L[i] = slab[tid + i * 256];

    float4 acc[4];
#pragma unroll
    for (int i = 0; i < 4; ++i) acc[i] = make_float4(0.f, 0.f, 0.f, 0.f);

    for (int d = 0; d < D_; ++d) {
        const float4* e4 = (const float4*)(emb + ((size_t)b * D_ + d) * HW_ + p0);
        if (d + 1 < D_) {
            __builtin_prefetch(emb + ((size_t)b * D_ + d + 1) * HW_ + p0 + tid * 16);
        }
#pragma unroll
        for (int i = 0; i < 4; ++i) {
            float4 v = e4[tid + i * 256];
            if (L[i].x > 0) { float t = v.x - lmean[d * K_ + L[i].x - 1]; acc[i].x += t * t; }
            if (L[i].y > 0) { float t = v.y - lmean[d * K_ + L[i].y - 1]; acc[i].y += t * t; }
            if (L[i].z > 0) { float t = v.z - lmean[d * K_ + L[i].z - 1]; acc[i].z += t * t; }
            if (L[i].w > 0) { float t = v.w - lmean[d * K_ + L[i].w - 1]; acc[i].w += t * t; }
        }
    }
#pragma unroll
    for (int i = 0; i < 4; ++i) {
        if (L[i].x > 0) { float t = sqrtf(acc[i].x) - 0.5f; if (t > 0.f) atomicAdd(&lvar[L[i].x - 1], t * t); }
        if (L[i].y > 0) { float t = sqrtf(acc[i].y) - 0.5f; if (t > 0.f) atomicAdd(&lvar[L[i].y - 1], t * t); }
        if (L[i].z > 0) { float t = sqrtf(acc[i].z) - 0.5f; if (t > 0.f) atomicAdd(&lvar[L[i].z - 1], t * t); }
        if (L[i].w > 0) { float t = sqrtf(acc[i].w) - 0.5f; if (t > 0.f) atomicAdd(&lvar[L[i].w - 1], t * t); }
    }
    __syncthreads();
    if (tid < K_) atomicAdd(&ws[WS_VAR + b * K_ + tid], lvar[tid]);
}

// Finalize: one wave per image. Gram of means via V_WMMA_F32_16X16X4_F32,
// then dist/var/reg terms reduced into d_out with float atomics.
__global__ void dl_finalize(const float* __restrict__ ws, float* __restrict__ out) {
    __shared__ float mM[B_][K_ * D_];  // k-major: mM[b][k*16 + d]
    __shared__ float n2[B_][K_];
    const int tid  = threadIdx.x;
    const int b    = tid >> 5;   // wave index = image
    const int lane = tid & 31;   // lane = cluster for the loads below

    float s2 = 0.f;
    for (int d = 0; d < D_; ++d) {
        float m = ws[WS_MEANS + b * D_ * K_ + d * K_ + lane];
        mM[b][lane * D_ + d] = m;
        s2 += m * m;
    }
    n2[b][lane] = s2;

    // reg term: mean_k ||mu_k|| averaged over images, * gamma
    atomicAdd(&out[2], sqrtf(s2) * (GAMMA_ / (float)(K_ * B_)));
    // var term
    float cnt = ws[WS_CNT + b * K_ + lane];
    float vv  = ws[WS_VAR + b * K_ + lane] / fmaxf(cnt, 1.f);
    atomicAdd(&out[0], vv * (1.f / (float)(K_ * B_)));

    __syncthreads();

    const int lr = lane & 15;   // column / row-within-half per WMMA layout
    const int hi = lane >> 4;   // 0: K {0,1} / rows +0;  1: K {2,3} / rows +8
    float dist_acc = 0.f;

#pragma unroll
    for (int I = 0; I < 2; ++I) {
#pragma unroll
        for (int J = 0; J < 2; ++J) {
            v8f cacc = {0.f, 0.f, 0.f, 0.f, 0.f, 0.f, 0.f, 0.f};
#pragma unroll
            for (int d0 = 0; d0 < D_; d0 += 4) {
                v2f av, bv;
                // A 16x4 frag: lane l -> row i0+lr, VGPR0/1 = K {0,1} (lanes 0-15) or {2,3} (16-31)
                av.x = mM[b][(I * 16 + lr) * D_ + d0 + 2 * hi + 0];
                av.y = mM[b][(I * 16 + lr) * D_ + d0 + 2 * hi + 1];
                // B 4x16 frag = transposed means tile; same per-lane gather with j0
                bv.x = mM[b][(J * 16 + lr) * D_ + d0 + 2 * hi + 0];
                bv.y = mM[b][(J * 16 + lr) * D_ + d0 + 2 * hi + 1];
                cacc = __builtin_amdgcn_wmma_f32_16x16x4_f32(
                    false, av, false, bv, (short)0, cacc, false, false);
            }
#pragma unroll
            for (int v = 0; v < 8; ++v) {
                int i = I * 16 + v + 8 * hi;  // C layout: VGPR v holds rows v / v+8
                int j = J * 16 + lr;
                if (i != j) {
                    float d2 = n2[b][i] + n2[b][j] - 2.f * cacc[v];
                    float dm = sqrtf(fmaxf(d2, 0.f));
                    float h  = fmaxf(3.0f - dm, 0.f);  // 2*delta_d = 3.0
                    dist_acc += h * h;
                }
            }
        }
    }
    atomicAdd(&out[1], dist_acc * (1.f / ((float)(K_ - 1) * (float)K_ * (float)B_)));
}

extern "C" void kernel_launch(void* const* d_in, const int* in_sizes, int n_in,
                              void* d_out, int out_size, void* d_ws, size_t ws_size,
                              hipStream_t stream) {
    const float* emb = (const float*)d_in[0];
    const int*   lab = (const int*)d_in[1];
    float* out = (float*)d_out;
    float* ws  = (float*)d_ws;

    dl_init<<<(WS_TOTAL + 255) / 256, 256, 0, stream>>>(ws, out);
    dl_sums<<<dim3(HW_ / 4096, B_), 256, 0, stream>>>(emb, lab, ws);
    dl_means<<<(B_ * D_ * K_ + 255) / 256, 256, 0, stream>>>(ws);
    dl_var<<<dim3(HW_ / 4096, B_), 256, 0, stream>>>(emb, lab, ws);
    dl_finalize<<<1, 256, 0, stream>>>(ws, out);

    (void)in_sizes; (void)n_in; (void)out_size; (void)ws_size;
}